// GPASelfAttention_29214367547803
// MI455X (gfx1250) — compile-verified
//
#include <hip/hip_runtime.h>
#include <math.h>

typedef float v2f __attribute__((ext_vector_type(2)));
typedef float v4f __attribute__((ext_vector_type(4)));
typedef float v8f __attribute__((ext_vector_type(8)));

#define B_      8
#define D_      4096
#define HQ_     32
#define HKV_    8
#define DH_     128
#define MAXSEQ_ 4096
#define NQ_     (HQ_*DH_)          // 4096
#define NKV_    (HKV_*DH_)         // 1024
#define NCAT_   (NQ_ + 2*NKV_)     // 6144
#define TCHUNK  128
#define NCHUNK  (MAXSEQ_/TCHUNK)   // 32
#define LDK     132                // padded LDS row stride (floats)

// ---------------------------------------------------------------------------
// Kernel 0: build a 16-row zero-padded copy of x (rows 8..15 = 0) so the
// WMMA A-operand loads need no EXEC-masked guard, and pre-zero the padded
// attention-output matrix (rows 8..15) used by the output projection.
// Grid: 256 blocks x 256 threads covers 16*4096 elements.
// ---------------------------------------------------------------------------
__global__ __launch_bounds__(256) void prep_kernel(
    const float* __restrict__ x, float* __restrict__ xpad,
    float* __restrict__ attnpad)
{
    const int i = blockIdx.x * 256 + threadIdx.x;    // 0 .. 65535
    const int r = i >> 12;                           // row 0..15
    xpad[i]    = (r < B_) ? x[i] : 0.f;
    attnpad[i] = 0.f;                                // rows 0..7 overwritten later
}

// ---------------------------------------------------------------------------
// Shared GEMM core: one wave computes a 16(M, 8 used) x 16(N) f32 tile over a
// 1024-deep K slice with V_WMMA_F32_16X16X4_F32.
//
// A (16x4, MxK): lane L holds row m = L&15; VGPR0/1 = K 2*(L>>4), 2*(L>>4)+1.
// B (4x16, KxN): lane L holds col n = L&15; VGPR0/1 = rows 2*(L>>4), +1.
// xmat must be the 16-row padded activation matrix (row stride D_).
// ---------------------------------------------------------------------------
__device__ __forceinline__ v8f wmma_colblock(const float* __restrict__ xmat,
                                             const float* __restrict__ W,
                                             int ldw, int ncol, int kBase, int lane)
{
    const int m    = lane & 15;
    const int psel = lane >> 4;          // 0 or 1
    v8f c = {0.f,0.f,0.f,0.f,0.f,0.f,0.f,0.f};
    const float* wp = W    + (size_t)(kBase + 2*psel) * ldw + (ncol + m);
    const float* xp = xmat + (size_t)m * D_ + (kBase + 2*psel);
    #pragma unroll 8
    for (int kk = 0; kk < 1024; kk += 4) {
        const v2f a = *(const v2f*)xp;          // unconditional b64 (padded rows)
        v2f b;
        b.x = wp[0];
        b.y = wp[ldw];
        __builtin_prefetch(wp + (size_t)8*ldw, 0, 3);   // WGP-scope prefetch
        c = __builtin_amdgcn_wmma_f32_16x16x4_f32(false, a, false, b,
                                                  (short)0, c, false, false);
        wp += (size_t)4*ldw;
        xp += 4;
    }
    return c;
}

// ---------------------------------------------------------------------------
// Kernel 1: fused QKV projection + bias + RoPE.
// Grid: NCAT_/16 = 384 blocks x 128 threads (4 waves, K split 4x1024).
// ---------------------------------------------------------------------------
__global__ __launch_bounds__(128) void qkv_rope_kernel(
    const float* __restrict__ xpad,
    const float* __restrict__ wq, const float* __restrict__ bq,
    const float* __restrict__ wk, const float* __restrict__ bk,
    const float* __restrict__ wv, const float* __restrict__ bv,
    const int*   __restrict__ start_pos_p,
    float* __restrict__ q_out, float* __restrict__ k_out, float* __restrict__ v_out)
{
    __shared__ float red[4][256];
    const int tid  = threadIdx.x;
    const int wave = tid >> 5;
    const int lane = tid & 31;
    const int n0   = blockIdx.x * 16;

    const float* W; const float* bias; int ldw; int ncol; int kind;
    if (n0 < NQ_)           { W = wq; bias = bq; ldw = NQ_;  ncol = n0;              kind = 0; }
    else if (n0 < NQ_+NKV_) { W = wk; bias = bk; ldw = NKV_; ncol = n0 - NQ_;        kind = 1; }
    else                    { W = wv; bias = bv; ldw = NKV_; ncol = n0 - NQ_ - NKV_; kind = 2; }

    v8f c = wmma_colblock(xpad, W, ldw, ncol, wave*1024, lane);

    // C layout: VGPR r, lanes 0-15 -> (M=r, N=lane); lanes 16-31 -> (M=r+8).
    #pragma unroll
    for (int r = 0; r < 8; ++r) red[wave][r*32 + lane] = c[r];
    __syncthreads();

    if (tid < 64) {
        const int mrow = tid >> 3;          // batch 0..7
        const int pr   = tid & 7;           // rope pair 0..7 within the 16-col tile
        const int ne = 2*pr, no = ne + 1;
        float ve = 0.f, vo = 0.f;
        #pragma unroll
        for (int w = 0; w < 4; ++w) { ve += red[w][mrow*32 + ne]; vo += red[w][mrow*32 + no]; }
        ve += bias[ncol + ne];
        vo += bias[ncol + no];
        if (kind != 2) {                    // RoPE on q and k only
            const int dhe  = (ncol + ne) & (DH_-1);
            const float pos = (float)start_pos_p[0];
            const float theta = powf(10000.0f, -(float)dhe * (1.0f/(float)DH_));
            const float ang = pos * theta;
            const float cs = cosf(ang), sn = sinf(ang);
            const float rr = ve*cs - vo*sn;
            const float ii = ve*sn + vo*cs;
            ve = rr; vo = ii;
        }
        const int gcol = ncol + ne;
        if (kind == 0) {
            q_out[(size_t)mrow*NQ_ + gcol]     = ve;
            q_out[(size_t)mrow*NQ_ + gcol + 1] = vo;
        } else if (kind == 1) {
            k_out[(size_t)mrow*NKV_ + gcol]     = ve;
            k_out[(size_t)mrow*NKV_ + gcol + 1] = vo;
        } else {
            v_out[(size_t)mrow*NKV_ + gcol]     = ve;
            v_out[(size_t)mrow*NKV_ + gcol + 1] = vo;
        }
    }
}

// ---------------------------------------------------------------------------
// Kernel 2: flash-decode partials. Grid: B*HKV*NCHUNK = 2048 blocks x 256 thr.
// Each block: 4 q-heads over a 128-deep t-chunk; keys/values staged coalesced
// through padded LDS; row t == start_pos is sourced from the fresh k/v
// (cache is read-only -> cache update virtualized).
// ---------------------------------------------------------------------------
__global__ __launch_bounds__(256) void attn_partial_kernel(
    const float* __restrict__ cache_k, const float* __restrict__ cache_v,
    const float* __restrict__ q_rope,  const float* __restrict__ k_new,
    const float* __restrict__ v_new,   const int* __restrict__ start_pos_p,
    float* __restrict__ partML, float* __restrict__ partO)
{
    __shared__ float kstage[64*LDK];    // 33 KB staging buffer (keys, then values)
    __shared__ float qs[4*DH_];
    __shared__ float pp[4*TCHUNK];
    __shared__ float redbuf[4*64];
    __shared__ float mlh[8];            // [0..3]=max, [4..7]=sumexp per head

    const int tid  = threadIdx.x;
    const int wg   = blockIdx.x;
    const int ch   = wg & (NCHUNK-1);
    const int kvh  = (wg >> 5) & (HKV_-1);
    const int bb   = wg >> 8;
    const int spos = start_pos_p[0];

    for (int i = tid; i < 4*DH_; i += 256)
        qs[i] = q_rope[((size_t)bb*HQ_ + kvh*4 + (i>>7))*DH_ + (i & (DH_-1))];

    const int hl = tid >> 6;            // head 0..3
    const int tl = tid & 63;            // local t 0..63

    // ---- scores ----
    for (int s = 0; s < 2; ++s) {
        __syncthreads();
        for (int i = tid; i < 64*(DH_/4); i += 256) {
            const int row  = i >> 5;
            const int col4 = (i & 31) << 2;
            const int tg   = ch*TCHUNK + s*64 + row;
            const float* src = (tg == spos)
                ? (k_new   + ((size_t)bb*HKV_ + kvh)*DH_ + col4)
                : (cache_k + (((size_t)bb*MAXSEQ_ + tg)*HKV_ + kvh)*DH_ + col4);
            const v4f val = *(const v4f*)src;
            float* dst = &kstage[row*LDK + col4];
            dst[0]=val.x; dst[1]=val.y; dst[2]=val.z; dst[3]=val.w;
        }
        __syncthreads();
        const float* kr = &kstage[tl*LDK];
        const float* qh = &qs[hl*DH_];
        float acc = 0.f;
        #pragma unroll
        for (int j = 0; j < DH_; j += 4) {
            const v4f kv_ = *(const v4f*)(kr + j);
            const v4f qv  = *(const v4f*)(qh + j);
            acc += qv.x*kv_.x + qv.y*kv_.y + qv.z*kv_.z + qv.w*kv_.w;
        }
        pp[hl*TCHUNK + s*64 + tl] = acc * 0.08838834764831845f; // 1/sqrt(128)
    }
    __syncthreads();

    // ---- per-head max / exp / sum ----
    const float v0 = pp[hl*TCHUNK + tl];
    const float v1 = pp[hl*TCHUNK + 64 + tl];
    redbuf[hl*64 + tl] = fmaxf(v0, v1);
    __syncthreads();
    for (int st2 = 32; st2 > 0; st2 >>= 1) {
        if (tl < st2) redbuf[hl*64+tl] = fmaxf(redbuf[hl*64+tl], redbuf[hl*64+tl+st2]);
        __syncthreads();
    }
    const float M = redbuf[hl*64];
    __syncthreads();
    const float e0 = __expf(v0 - M), e1 = __expf(v1 - M);
    pp[hl*TCHUNK + tl]      = e0;
    pp[hl*TCHUNK + 64 + tl] = e1;
    redbuf[hl*64 + tl] = e0 + e1;
    __syncthreads();
    for (int st2 = 32; st2 > 0; st2 >>= 1) {
        if (tl < st2) redbuf[hl*64+tl] += redbuf[hl*64+tl+st2];
        __syncthreads();
    }
    if (tl == 0) { mlh[hl] = M; mlh[4+hl] = redbuf[hl*64]; }

    // ---- p @ V partial: 512 outputs (4 heads x 128 dh), 2 per thread ----
    const int h0  = tid >> 7;           // 0..1
    const int dh0 = tid & (DH_-1);
    const int h1  = h0 + 2;
    float o0 = 0.f, o1 = 0.f;
    for (int s = 0; s < 2; ++s) {
        __syncthreads();
        for (int i = tid; i < 64*(DH_/4); i += 256) {
            const int row  = i >> 5;
            const int col4 = (i & 31) << 2;
            const int tg   = ch*TCHUNK + s*64 + row;
            const float* src = (tg == spos)
                ? (v_new   + ((size_t)bb*HKV_ + kvh)*DH_ + col4)
                : (cache_v + (((size_t)bb*MAXSEQ_ + tg)*HKV_ + kvh)*DH_ + col4);
            const v4f val = *(const v4f*)src;
            float* dst = &kstage[row*LDK + col4];
            dst[0]=val.x; dst[1]=val.y; dst[2]=val.z; dst[3]=val.w;
        }
        __syncthreads();
        const float* p0 = &pp[h0*TCHUNK + s*64];
        const float* p1 = &pp[h1*TCHUNK + s*64];
        #pragma unroll 4
        for (int t = 0; t < 64; ++t) {
            const float vv = kstage[t*LDK + dh0];
            o0 += p0[t]*vv;
            o1 += p1[t]*vv;
        }
    }
    __syncthreads();
    if (tid < 8) partML[(size_t)wg*8 + tid] = mlh[tid];
    partO[(size_t)wg*512 + tid]       = o0;
    partO[(size_t)wg*512 + 256 + tid] = o1;
}

// ---------------------------------------------------------------------------
// Kernel 3: log-sum-exp combine over the 32 chunks. Grid: B*HKV = 64 x 256.
// Writes into the 16-row padded attention matrix (rows 0..7).
// ---------------------------------------------------------------------------
__global__ __launch_bounds__(256) void attn_combine_kernel(
    const float* __restrict__ partML, const float* __restrict__ partO,
    float* __restrict__ attnpad)
{
    const int tid = threadIdx.x;
    const int kvh = blockIdx.x & (HKV_-1);
    const int bb  = blockIdx.x >> 3;
    const int base = (bb*HKV_ + kvh) * NCHUNK;

    const int h0 = tid >> 7;
    const int dh = tid & (DH_-1);
    const int h1 = h0 + 2;

    float M0 = -3.0e38f, M1 = -3.0e38f;
    for (int c2 = 0; c2 < NCHUNK; ++c2) {
        const float* ml = partML + (size_t)(base + c2)*8;
        M0 = fmaxf(M0, ml[h0]);
        M1 = fmaxf(M1, ml[h1]);
    }
    float L0 = 0.f, L1 = 0.f, O0 = 0.f, O1 = 0.f;
    for (int c2 = 0; c2 < NCHUNK; ++c2) {
        const float* ml = partML + (size_t)(base + c2)*8;
        const float* po = partO  + (size_t)(base + c2)*512;
        const float w0 = __expf(ml[h0] - M0);
        const float w1 = __expf(ml[h1] - M1);
        L0 += ml[4+h0]*w0;
        L1 += ml[4+h1]*w1;
        O0 += po[h0*DH_ + dh]*w0;
        O1 += po[h1*DH_ + dh]*w1;
    }
    attnpad[(size_t)bb*D_ + (kvh*4 + h0)*DH_ + dh] = O0 / L0;
    attnpad[(size_t)bb*D_ + (kvh*4 + h1)*DH_ + dh] = O1 / L1;
}

// ---------------------------------------------------------------------------
// Kernel 4: output projection (attnpad[16][4096] @ wo[4096][4096] + bo).
// Grid: 4096/16 = 256 blocks x 128 threads.
// ---------------------------------------------------------------------------
__global__ __launch_bounds__(128) void out_proj_kernel(
    const float* __restrict__ attnpad, const float* __restrict__ wo,
    const float* __restrict__ bo,      float* __restrict__ out)
{
    __shared__ float red[4][256];
    const int tid  = threadIdx.x;
    const int wave = tid >> 5;
    const int lane = tid & 31;
    const int n0   = blockIdx.x * 16;

    v8f c = wmma_colblock(attnpad, wo, D_, n0, wave*1024, lane);
    #pragma unroll
    for (int r = 0; r < 8; ++r) red[wave][r*32 + lane] = c[r];
    __syncthreads();

    const int mrow = tid >> 4;   // batch 0..7
    const int nn   = tid & 15;
    float v = bo[n0 + nn];
    #pragma unroll
    for (int w = 0; w < 4; ++w) v += red[w][mrow*32 + nn];
    out[(size_t)mrow*D_ + n0 + nn] = v;
}

// ---------------------------------------------------------------------------
extern "C" void kernel_launch(void* const* d_in, const int* in_sizes, int n_in,
                              void* d_out, int out_size, void* d_ws, size_t ws_size,
                              hipStream_t stream)
{
    const float* x        = (const float*)d_in[0];
    const float* wq       = (const float*)d_in[1];
    const float* bq       = (const float*)d_in[2];
    const float* wk       = (const float*)d_in[3];
    const float* bk       = (const float*)d_in[4];
    const float* wv       = (const float*)d_in[5];
    const float* bv       = (const float*)d_in[6];
    const float* wo       = (const float*)d_in[7];
    const float* bo       = (const float*)d_in[8];
    const float* cache_k  = (const float*)d_in[9];
    const float* cache_v  = (const float*)d_in[10];
    const int*   start_pos= (const int*)d_in[11];
    float* out = (float*)d_out;

    // workspace layout (floats): ~5 MB total
    float* ws       = (float*)d_ws;
    float* q_rope   = ws;                  // 8*32*128   = 32768
    float* k_new    = ws + 32768;          // 8*8*128    =  8192
    float* v_new    = ws + 40960;          //            =  8192
    float* xpad     = ws + 49152;          // 16*4096    = 65536
    float* attnpad  = ws + 114688;         // 16*4096    = 65536
    float* partML   = ws + 180224;         // 2048*8     = 16384
    float* partO    = ws + 196608;         // 2048*512   = 1048576

    prep_kernel<<<256, 256, 0, stream>>>(x, xpad, attnpad);
    qkv_rope_kernel<<<NCAT_/16, 128, 0, stream>>>(xpad, wq, bq, wk, bk, wv, bv,
                                                  start_pos, q_rope, k_new, v_new);
    attn_partial_kernel<<<B_*HKV_*NCHUNK, 256, 0, stream>>>(cache_k, cache_v,
                                                            q_rope, k_new, v_new,
                                                            start_pos, partML, partO);
    attn_combine_kernel<<<B_*HKV_, 256, 0, stream>>>(partML, partO, attnpad);
    out_proj_kernel<<<D_/16, 128, 0, stream>>>(attnpad, wo, bo, out);
}